// Model_66168266162562
// MI455X (gfx1250) — compile-verified
//
#include <hip/hip_runtime.h>

// GCN: h1 = relu(Norm(A+I) @ (X@W1) + b1); h2 = Norm(A+I) @ (h1@W2) + b2;
// score[e] = dot(h2[src[e]], h2[dst[e]])
// N=50000, E=800000, DIN=128, DH=128, DOUT=64

#define NDIM_IN  128
#define NDIM_HID 128
#define NDIM_OUT 64

typedef __attribute__((ext_vector_type(2))) float v2f;
typedef __attribute__((ext_vector_type(8))) float v8f;

// ---------------- degree / normalization ----------------

__global__ void k_init_deg(float* __restrict__ deg, int n) {
    int i = blockIdx.x * blockDim.x + threadIdx.x;
    if (i < n) deg[i] = 1.0f;  // self-loop contributes 1
}

__global__ void k_count_deg(const int* __restrict__ dst, float* __restrict__ deg, int e) {
    int i = blockIdx.x * blockDim.x + threadIdx.x;
    if (i < e) atomicAdd(&deg[dst[i]], 1.0f);
}

__global__ void k_deg_to_dinv(float* __restrict__ deg, int n) {
    int i = blockIdx.x * blockDim.x + threadIdx.x;
    if (i < n) deg[i] = rsqrtf(deg[i]);  // deg >= 1 always (self-loops)
}

__global__ void k_edge_coef(const int* __restrict__ src, const int* __restrict__ dst,
                            const float* __restrict__ dinv, float* __restrict__ coef, int e) {
    int i = blockIdx.x * blockDim.x + threadIdx.x;
    if (i < e) coef[i] = dinv[src[i]] * dinv[dst[i]];
}

// ---------------- fp32 WMMA GEMM: C[M,Ndim] = A[M,Kdim] @ B[Kdim,Ndim] ----------------
// One wave per 16x16 output tile; K-loop in steps of 4 using V_WMMA_F32_16X16X4_F32.
// A-frag (16x4 f32, 2 VGPR/lane): lanes 0-15 hold M=lane K={0,1}; lanes 16-31 M=lane-16 K={2,3}.
// B-frag (4x16 f32, 2 VGPR/lane): v0 = row K (lanes<16: K+0, lanes>=16: K+2), v1 = next row.
// C/D (8 VGPR): VGPR r -> M = r + 8*(lane>=16), N = lane&15.
template<int Kdim, int Ndim>
__global__ __launch_bounds__(256) void k_wmma_gemm(const float* __restrict__ A,
                                                   const float* __restrict__ B,
                                                   float* __restrict__ C, int M) {
    static_assert(Kdim % 4 == 0 && Ndim % 16 == 0, "tile shapes");
    const int lane = threadIdx.x & 31;
    const int wave = threadIdx.x >> 5;
    const int tilesN = Ndim / 16;
    const int numMt = M / 16;                      // M = 50000 -> 3125 exact
    const int w = blockIdx.x * 8 + wave;
    const int mt = w / tilesN;
    const int nt = w % tilesN;
    if (mt >= numMt) return;                       // wave-uniform: EXEC all-1s inside

    const int half = lane >> 4;                    // 0 or 1
    const int lm   = lane & 15;
    const int row  = mt * 16 + lm;                 // A row this lane supplies
    const int col  = nt * 16 + lm;                 // B/C column this lane supplies

    v8f acc = {};
    const float* Arow = A + (long long)row * Kdim + 2 * half;  // K offset 0 or 2
    const float* Bcol = B + col + (long long)(2 * half) * Ndim;

#pragma unroll 4
    for (int k = 0; k < Kdim; k += 4) {
        v2f a = *(const v2f*)(Arow + k);           // 8B-aligned (k,half even offsets)
        v2f b;
        b.x = Bcol[(long long)k * Ndim];
        b.y = Bcol[(long long)(k + 1) * Ndim];
        acc = __builtin_amdgcn_wmma_f32_16x16x4_f32(
            /*neg_a=*/false, a, /*neg_b=*/false, b,
            /*c_mod=*/(short)0, acc, /*reuse_a=*/false, /*reuse_b=*/false);
    }

    float* Crow = C + (long long)(mt * 16 + 8 * half) * Ndim + col;
#pragma unroll
    for (int r = 0; r < 8; ++r) Crow[(long long)r * Ndim] = acc[r];
}

// ---------------- aggregation ----------------

// out[v][j] = xw[v][j] * dinv[v]^2   (self-loop term seeds the accumulator)
template<int D>
__global__ void k_self_loop_init(const float* __restrict__ xw, const float* __restrict__ dinv,
                                 float* __restrict__ out, int n) {
    int tid = blockIdx.x * blockDim.x + threadIdx.x;
    if (tid < n * D) {
        float di = dinv[tid / D];
        out[tid] = xw[tid] * di * di;
    }
}

// out[dst[e]][j] += xw[src[e]][j] * coef[e]
template<int D>
__global__ void k_edge_scatter(const float* __restrict__ xw, const int* __restrict__ src,
                               const int* __restrict__ dst, const float* __restrict__ coef,
                               float* __restrict__ out, int e) {
    static_assert((D & (D - 1)) == 0, "D pow2");
    int tid = blockIdx.x * blockDim.x + threadIdx.x;
    if (tid >= e * D) return;
    int ei = tid / D;
    int j  = tid & (D - 1);
    float v = xw[(long long)src[ei] * D + j] * coef[ei];
    atomicAdd(&out[(long long)dst[ei] * D + j], v);
}

template<int D, bool RELU>
__global__ void k_bias_act(float* __restrict__ h, const float* __restrict__ b, int n) {
    int tid = blockIdx.x * blockDim.x + threadIdx.x;
    if (tid < n * D) {
        float v = h[tid] + b[tid & (D - 1)];
        h[tid] = RELU ? fmaxf(v, 0.0f) : v;
    }
}

// ---------------- edge scoring: one wave32 per edge ----------------

__global__ __launch_bounds__(256) void k_edge_score(const float* __restrict__ h,
                                                    const int* __restrict__ src,
                                                    const int* __restrict__ dst,
                                                    float* __restrict__ score, int e) {
    int lane = threadIdx.x & 31;
    int ei = (blockIdx.x * blockDim.x + threadIdx.x) >> 5;
    if (ei >= e) return;
    const float* a = h + (long long)src[ei] * NDIM_OUT;
    const float* b = h + (long long)dst[ei] * NDIM_OUT;
    float s = a[lane] * b[lane] + a[lane + 32] * b[lane + 32];
#pragma unroll
    for (int off = 16; off > 0; off >>= 1) s += __shfl_xor(s, off, 32);
    if (lane == 0) score[ei] = s;
}

// ---------------- launch ----------------

extern "C" void kernel_launch(void* const* d_in, const int* in_sizes, int n_in,
                              void* d_out, int out_size, void* d_ws, size_t ws_size,
                              hipStream_t stream) {
    const float* X    = (const float*)d_in[0];
    const int*   esrc = (const int*)d_in[1];
    const int*   edst = (const int*)d_in[2];
    const float* W1   = (const float*)d_in[3];
    const float* b1   = (const float*)d_in[4];
    const float* W2   = (const float*)d_in[5];
    const float* b2   = (const float*)d_in[6];
    float* out = (float*)d_out;

    const int n = in_sizes[0] / NDIM_IN;   // 50000
    const int e = in_sizes[1];             // 800000

    float* ws   = (float*)d_ws;
    float* dinv = ws;                            // n floats
    float* coef = dinv + n;                      // e floats
    float* bufA = coef + e;                      // n*128 floats
    float* bufB = bufA + (size_t)n * NDIM_HID;   // n*128 floats

    const int T = 256;
    const int numMt = n / 16;

    // normalization
    k_init_deg   <<<(n + T - 1) / T, T, 0, stream>>>(dinv, n);
    k_count_deg  <<<(e + T - 1) / T, T, 0, stream>>>(edst, dinv, e);
    k_deg_to_dinv<<<(n + T - 1) / T, T, 0, stream>>>(dinv, n);
    k_edge_coef  <<<(e + T - 1) / T, T, 0, stream>>>(esrc, edst, dinv, coef, e);

    // layer 1: XW1 -> bufA; aggregate -> bufB; bias+relu in place
    {
        int tiles = numMt * (NDIM_HID / 16);
        k_wmma_gemm<NDIM_IN, NDIM_HID><<<(tiles + 7) / 8, T, 0, stream>>>(X, W1, bufA, n);
        int nd = n * NDIM_HID, ed = e * NDIM_HID;
        k_self_loop_init<NDIM_HID><<<(nd + T - 1) / T, T, 0, stream>>>(bufA, dinv, bufB, n);
        k_edge_scatter<NDIM_HID><<<(ed + T - 1) / T, T, 0, stream>>>(bufA, esrc, edst, coef, bufB, e);
        k_bias_act<NDIM_HID, true><<<(nd + T - 1) / T, T, 0, stream>>>(bufB, b1, n);
    }

    // layer 2: bufB@W2 -> bufA (first n*64); aggregate -> bufB (first n*64); bias in place
    {
        int tiles = numMt * (NDIM_OUT / 16);
        k_wmma_gemm<NDIM_HID, NDIM_OUT><<<(tiles + 7) / 8, T, 0, stream>>>(bufB, W2, bufA, n);
        int nd = n * NDIM_OUT, ed = e * NDIM_OUT;
        k_self_loop_init<NDIM_OUT><<<(nd + T - 1) / T, T, 0, stream>>>(bufA, dinv, bufB, n);
        k_edge_scatter<NDIM_OUT><<<(ed + T - 1) / T, T, 0, stream>>>(bufA, esrc, edst, coef, bufB, e);
        k_bias_act<NDIM_OUT, false><<<(nd + T - 1) / T, T, 0, stream>>>(bufB, b2, n);
    }

    // per-edge dot product of endpoint embeddings
    k_edge_score<<<(e * 32 + T - 1) / T, T, 0, stream>>>(bufB, esrc, edst, out, e);
}